// Encoder_2241972929027
// MI455X (gfx1250) — compile-verified
//
#include <hip/hip_runtime.h>
#include <hip/hip_bf16.h>

// ---- problem constants (match reference) ----
#define N_PART   4000
#define BATCH    2
#define NROWS    8000            // B*N
#define RRAD     0.05f
#define RSQ      (RRAD*RRAD)
#define MAXE     400000
#define LNEPS    1e-5f
#define NF_STRIDE 32             // node features padded 30 -> 32
#define HSTR     132             // LDS activation stride (128 + pad, kills bank conflicts)
#define WPB      4               // waves (16-row tiles) per block

typedef float v2f __attribute__((ext_vector_type(2)));
typedef float v8f __attribute__((ext_vector_type(8)));

#define LDS_FENCE() asm volatile("s_wait_dscnt 0" ::: "memory")

__device__ __forceinline__ v8f wmma_f32(v2f a, v2f b, v8f c) {
  // V_WMMA_F32_16X16X4_F32 : D = A(16x4) * B(4x16) + C(16x16), full f32
  return __builtin_amdgcn_wmma_f32_16x16x4_f32(false, a, false, b, (short)0, c, false, false);
}

struct alignas(16) WaveScratch {
  float h[16 * HSTR];   // 16x128 activation tile (padded stride)
  float mu[16];
  float rs[16];
  float vf[16];
};

// ---- store a 16x128 accumulator tile (8 x v8f) into the LDS activation buffer ----
template <bool RELU>
__device__ __forceinline__ void store_acc(const v8f* acc, float* h, int lo, int hi) {
  LDS_FENCE();
#pragma unroll
  for (int t = 0; t < 8; ++t) {
    int col = t * 16 + lo;
#pragma unroll
    for (int v = 0; v < 8; ++v) {
      float x = acc[t][v];
      if (RELU) x = fmaxf(x, 0.f);
      h[(v + 8 * hi) * HSTR + col] = x;   // C layout: row = v + 8*hi, col = t*16 + lo
    }
  }
  LDS_FENCE();
}

// ---- 128 -> 128 dense layer: A from LDS h, B from packed weights (one b64 per WMMA) ----
// Wp packed layout: Wp2[(k/2)*128 + col] = { W[k][col], W[k+1][col] } for even k
template <bool RELU>
__device__ __forceinline__ void layer128(const v2f* __restrict__ Wp2,
                                         const float* __restrict__ bias,
                                         float* h, int lo, int hi) {
  v8f acc[8];
#pragma unroll
  for (int t = 0; t < 8; ++t) {
    float bv = bias[t * 16 + lo];
#pragma unroll
    for (int v = 0; v < 8; ++v) acc[t][v] = bv;
  }
#pragma unroll 2
  for (int k0 = 0; k0 < 128; k0 += 4) {
    int ka = k0 + 2 * hi;                 // A frag: lanes0-15 K=k0..k0+1, lanes16-31 K=k0+2..k0+3
    v2f a = *(const v2f*)(h + lo * HSTR + ka);
    const v2f* wrow = Wp2 + (ka >> 1) * 128;
#pragma unroll
    for (int t = 0; t < 8; ++t) {
      int col = t * 16 + lo;
      acc[t] = wmma_f32(a, wrow[col], acc[t]);
    }
  }
  store_acc<RELU>(acc, h, lo, hi);
}

// ---- LayerNorm over 128 cols of the 16-row LDS tile, vectorized coalesced store ----
__device__ __forceinline__ void layernorm_store(WaveScratch& S,
                                                const float* __restrict__ g,
                                                const float* __restrict__ bta,
                                                bool use_vf,
                                                float* __restrict__ outbase,
                                                int lo, int hi, int lane) {
  if (hi == 0) {                       // lanes 0..15: one row each, float4 reads (bank-conflict-free)
    const float4* hr = (const float4*)(S.h + lo * HSTR);
    float s = 0.f, ss = 0.f;
#pragma unroll 4
    for (int c = 0; c < 32; ++c) {
      float4 x = hr[c];
      s  += x.x + x.y + x.z + x.w;
      ss += x.x * x.x + x.y * x.y + x.z * x.z + x.w * x.w;
    }
    float m = s * (1.f / 128.f);
    float var = ss * (1.f / 128.f) - m * m;
    S.mu[lo] = m;
    S.rs[lo] = rsqrtf(var + LNEPS);
  }
  LDS_FENCE();
  int col = lane * 4;                  // each lane owns 4 consecutive cols; row-burst stores
  float4 gg = *(const float4*)(g + col);
  float4 bb = *(const float4*)(bta + col);
#pragma unroll 4
  for (int row = 0; row < 16; ++row) {
    float4 x = *(const float4*)(S.h + row * HSTR + col);
    float m = S.mu[row], r = S.rs[row];
    float vfm = use_vf ? S.vf[row] : 1.f;
    float4 o;
    o.x = ((x.x - m) * r * gg.x + bb.x) * vfm;
    o.y = ((x.y - m) * r * gg.y + bb.y) * vfm;
    o.z = ((x.z - m) * r * gg.z + bb.z) * vfm;
    o.w = ((x.w - m) * r * gg.w + bb.w) * vfm;
    *(float4*)(outbase + row * 128 + col) = o;   // 512B contiguous per row across the wave
  }
}

// =====================  kernel 0: weight fragment packing  =====================
// Wp[i*2 + {0,1}] = { W[k][col], W[k+1][col] } with zero-pad past Kreal rows.
__global__ void pack_w_kernel(const float* __restrict__ W, int Kreal, int Kpad,
                              float* __restrict__ Wp) {
  int i = blockIdx.x * blockDim.x + threadIdx.x;     // over (Kpad/2)*128 pairs
  int tot = (Kpad >> 1) * 128;
  if (i >= tot) return;
  int kh = i >> 7, col = i & 127;
  int k = kh * 2;
  Wp[i * 2 + 0] = (k     < Kreal) ? W[k * 128 + col]       : 0.f;
  Wp[i * 2 + 1] = (k + 1 < Kreal) ? W[(k + 1) * 128 + col] : 0.f;
}

// =====================  kernel 1: node features  =====================
__global__ void node_feats_kernel(const float* __restrict__ pos, const float* __restrict__ vel,
                                  const int* __restrict__ mat,
                                  const float* __restrict__ vmean, const float* __restrict__ vstd,
                                  const float* __restrict__ matW, const float* __restrict__ matb,
                                  float* __restrict__ nf) {
  int gidx = blockIdx.x * blockDim.x + threadIdx.x;
  if (gidx >= NROWS) return;
  float m0 = vmean[0], m1 = vmean[1];
  float is0 = 1.f / vstd[0], is1 = 1.f / vstd[1];
  float* o = nf + (size_t)gidx * NF_STRIDE;
#pragma unroll
  for (int v = 0; v < 5; ++v) {
    o[v * 2 + 0] = (vel[(gidx * 5 + v) * 2 + 0] - m0) * is0;
    o[v * 2 + 1] = (vel[(gidx * 5 + v) * 2 + 1] - m1) * is1;
  }
  int mi = mat[gidx];
#pragma unroll
  for (int j = 0; j < 16; ++j) o[10 + j] = matW[mi * 16 + j] + matb[j];
  float px = pos[gidx * 2], py = pos[gidx * 2 + 1];
  float inv = 1.f / RRAD;
  o[26] = fminf(fmaxf((px - 0.f) * inv, -1.f), 1.f);
  o[27] = fminf(fmaxf((1.f - px) * inv, -1.f), 1.f);
  o[28] = fminf(fmaxf((py - 0.f) * inv, -1.f), 1.f);
  o[29] = fminf(fmaxf((1.f - py) * inv, -1.f), 1.f);
  o[30] = 0.f; o[31] = 0.f;
}

// =====================  kernel 2: per-row neighbor counts  =====================
__global__ void rowcount_kernel(const float* __restrict__ pos, int* __restrict__ rc) {
  int wave = threadIdx.x >> 5, lane = threadIdx.x & 31;
  int rg = blockIdx.x * 4 + wave;
  if (rg >= NROWS) return;
  int b = rg / N_PART;
  float px = pos[rg * 2], py = pos[rg * 2 + 1];
  const float* pb = pos + (size_t)b * N_PART * 2;
  int cnt = 0;
  for (int s0 = 0; s0 < N_PART; s0 += 32) {
    int s = s0 + lane;
    bool pred = false;
    if (s < N_PART) {
      float dx = px - pb[s * 2], dy = py - pb[s * 2 + 1];
      pred = (dx * dx + dy * dy) < RSQ;
    }
    unsigned m = (unsigned)__ballot(pred);
    cnt += __popc(m);
  }
  if (lane == 0) rc[rg] = cnt;
}

// =====================  kernel 3: exclusive scan (single block)  =====================
__global__ void scan_kernel(const int* __restrict__ rc, int* __restrict__ rs, int* __restrict__ total) {
  __shared__ int ps[1024];
  int t = threadIdx.x;
  int base = t * 8;
  int loc[8]; int s = 0;
#pragma unroll
  for (int i = 0; i < 8; ++i) {
    int x = (base + i) < NROWS ? rc[base + i] : 0;
    loc[i] = s; s += x;
  }
  ps[t] = s;
  __syncthreads();
  for (int off = 1; off < 1024; off <<= 1) {
    int add = (t >= off) ? ps[t - off] : 0;
    __syncthreads();
    ps[t] += add;
    __syncthreads();
  }
  int excl = (t == 0) ? 0 : ps[t - 1];
#pragma unroll
  for (int i = 0; i < 8; ++i)
    if (base + i < NROWS) rs[base + i] = excl + loc[i];
  if (t == 1023) *total = ps[1023];
}

// =====================  kernel 4: ordered edge emit  =====================
__global__ void emit_kernel(const float* __restrict__ pos, const int* __restrict__ rs,
                            int* __restrict__ eidx) {
  int wave = threadIdx.x >> 5, lane = threadIdx.x & 31;
  int rg = blockIdx.x * 4 + wave;
  if (rg >= NROWS) return;
  int b = rg / N_PART, r = rg % N_PART;
  float px = pos[rg * 2], py = pos[rg * 2 + 1];
  const float* pb = pos + (size_t)b * N_PART * 2;
  int base = rs[rg];
  unsigned lanemask = (lane == 0) ? 0u : ((1u << lane) - 1u);
  for (int s0 = 0; s0 < N_PART; s0 += 32) {
    int s = s0 + lane;
    bool pred = false;
    if (s < N_PART) {
      float dx = px - pb[s * 2], dy = py - pb[s * 2 + 1];
      pred = (dx * dx + dy * dy) < RSQ;
    }
    unsigned m = (unsigned)__ballot(pred);
    if (pred) {
      int e = base + __popc(m & lanemask);
      if (e < MAXE) {
        eidx[e * 3 + 0] = b;
        eidx[e * 3 + 1] = r;
        eidx[e * 3 + 2] = s;
      }
    }
    base += __popc(m);
  }
}

// =====================  kernel 5: finalize idx / valid outputs  =====================
__global__ void finalize_kernel(int* __restrict__ eidx, const int* __restrict__ nedge,
                                float* __restrict__ out_nidx, float* __restrict__ out_valid) {
  int e = blockIdx.x * blockDim.x + threadIdx.x;
  if (e >= MAXE) return;
  int ne = *nedge;
  bool valid = e < ne;
  out_valid[e] = valid ? 1.f : 0.f;
  if (valid) {
    out_nidx[(size_t)e * 3 + 0] = (float)eidx[e * 3 + 0];
    out_nidx[(size_t)e * 3 + 1] = (float)eidx[e * 3 + 1];
    out_nidx[(size_t)e * 3 + 2] = (float)eidx[e * 3 + 2];
  } else {
    eidx[e * 3 + 0] = 0; eidx[e * 3 + 1] = 0; eidx[e * 3 + 2] = 0;
    out_nidx[(size_t)e * 3 + 0] = 0.f;
    out_nidx[(size_t)e * 3 + 1] = 0.f;
    out_nidx[(size_t)e * 3 + 2] = 0.f;
  }
}

__global__ void mask_kernel(float* __restrict__ out_mask) {
  int i = blockIdx.x * blockDim.x + threadIdx.x;
  if (i < NROWS) out_mask[i] = 1.f;
}

// =====================  kernel 6: fused node MLP + LN (WMMA f32)  =====================
__global__ void __launch_bounds__(WPB * 32)
node_mlp_kernel(const float* __restrict__ nf,
                const float* __restrict__ W1p, const float* __restrict__ b1,
                const float* __restrict__ W2p, const float* __restrict__ b2,
                const float* __restrict__ W3p, const float* __restrict__ b3,
                const float* __restrict__ g, const float* __restrict__ bta,
                float* __restrict__ out_nodes) {
  __shared__ WaveScratch sws[WPB];
  int wave = threadIdx.x >> 5, lane = threadIdx.x & 31;
  int lo = lane & 15, hi = lane >> 4;
  WaveScratch& S = sws[wave];
  int tile = blockIdx.x * WPB + wave;         // 500 tiles of 16 rows, exact
  int row = tile * 16 + lo;
  const v2f* W1p2 = (const v2f*)W1p;

  // ---- layer 1: K = 32 (features padded; packed W1 has zero rows 30..31) ----
  v8f acc[8];
#pragma unroll
  for (int t = 0; t < 8; ++t) {
    float bv = b1[t * 16 + lo];
#pragma unroll
    for (int v = 0; v < 8; ++v) acc[t][v] = bv;
  }
#pragma unroll
  for (int k0 = 0; k0 < 32; k0 += 4) {
    int ka = k0 + 2 * hi;
    v2f a = *(const v2f*)(nf + (size_t)row * NF_STRIDE + ka);
    const v2f* wrow = W1p2 + (ka >> 1) * 128;
#pragma unroll
    for (int t = 0; t < 8; ++t) {
      int col = t * 16 + lo;
      acc[t] = wmma_f32(a, wrow[col], acc[t]);
    }
  }
  store_acc<true>(acc, S.h, lo, hi);

  layer128<true >((const v2f*)W2p, b2, S.h, lo, hi);
  layer128<false>((const v2f*)W3p, b3, S.h, lo, hi);
  layernorm_store(S, g, bta, /*use_vf=*/false,
                  out_nodes + (size_t)tile * 16 * 128, lo, hi, lane);
}

// =====================  kernel 7: fused edge MLP + LN (WMMA f32)  =====================
__global__ void __launch_bounds__(WPB * 32)
edge_mlp_kernel(const float* __restrict__ pos, const int* __restrict__ eidx,
                const int* __restrict__ nedge,
                const float* __restrict__ W1p, const float* __restrict__ b1,
                const float* __restrict__ W2p, const float* __restrict__ b2,
                const float* __restrict__ W3p, const float* __restrict__ b3,
                const float* __restrict__ g, const float* __restrict__ bta,
                float* __restrict__ out_edges) {
  __shared__ WaveScratch sws[WPB];
  int wave = threadIdx.x >> 5, lane = threadIdx.x & 31;
  int lo = lane & 15, hi = lane >> 4;
  WaveScratch& S = sws[wave];
  int tile = blockIdx.x * WPB + wave;         // 25000 tiles of 16 edges, exact
  int e0 = tile * 16;
  int ne = *nedge;

  // per-lane edge features for row m = lo (both lane halves compute the same row)
  int e = e0 + lo;
  float f0 = 0.f, f1 = 0.f, f2 = 0.f, f3 = 0.f;
  if (e < ne) {
    int bi = eidx[e * 3 + 0];
    int ri = eidx[e * 3 + 1];
    int si = eidx[e * 3 + 2];
    const float* pb = pos + (size_t)bi * N_PART * 2;
    float dx = (pb[ri * 2 + 0] - pb[si * 2 + 0]) * (1.f / RRAD);
    float dy = (pb[ri * 2 + 1] - pb[si * 2 + 1]) * (1.f / RRAD);
    float d2 = dx * dx + dy * dy;
    f0 = dx; f1 = dy;
    f2 = (d2 > 0.f) ? sqrtf(d2) : 0.f;
  }
  if (lane < 16) S.vf[lane] = ((e0 + lane) < ne) ? 1.f : 0.f;

  // ---- layer 1: K = 4 (3 feats + zero pad; packed W1 has zero row 3) ----
  v8f acc[8];
#pragma unroll
  for (int t = 0; t < 8; ++t) {
    float bv = b1[t * 16 + lo];
#pragma unroll
    for (int v = 0; v < 8; ++v) acc[t][v] = bv;
  }
  v2f a;
  a.x = hi ? f2 : f0;
  a.y = hi ? f3 : f1;
  const v2f* wrow = (const v2f*)W1p + hi * 128;   // (ka>>1) with ka = 2*hi
#pragma unroll
  for (int t = 0; t < 8; ++t) {
    int col = t * 16 + lo;
    acc[t] = wmma_f32(a, wrow[col], acc[t]);
  }
  store_acc<true>(acc, S.h, lo, hi);

  layer128<true >((const v2f*)W2p, b2, S.h, lo, hi);
  layer128<false>((const v2f*)W3p, b3, S.h, lo, hi);
  layernorm_store(S, g, bta, /*use_vf=*/true,
                  out_edges + (size_t)e0 * 128, lo, hi, lane);
}

// =====================  host launch  =====================
extern "C" void kernel_launch(void* const* d_in, const int* in_sizes, int n_in,
                              void* d_out, int out_size, void* d_ws, size_t ws_size,
                              hipStream_t stream) {
  (void)in_sizes; (void)n_in; (void)out_size; (void)ws_size;

  const float* pos   = (const float*)d_in[0];
  const float* vel   = (const float*)d_in[1];
  const int*   mat   = (const int*)d_in[2];
  // d_in[3] node_mask: all-true, not needed for compute
  const float* vmean = (const float*)d_in[4];
  const float* vstd  = (const float*)d_in[5];
  const float* matW  = (const float*)d_in[6];
  const float* matb  = (const float*)d_in[7];
  const float* nW1 = (const float*)d_in[8],  *nb1 = (const float*)d_in[9];
  const float* nW2 = (const float*)d_in[10], *nb2 = (const float*)d_in[11];
  const float* nW3 = (const float*)d_in[12], *nb3 = (const float*)d_in[13];
  const float* nlg = (const float*)d_in[14], *nlb = (const float*)d_in[15];
  const float* eW1 = (const float*)d_in[16], *eb1 = (const float*)d_in[17];
  const float* eW2 = (const float*)d_in[18], *eb2 = (const float*)d_in[19];
  const float* eW3 = (const float*)d_in[20], *eb3 = (const float*)d_in[21];
  const float* elg = (const float*)d_in[22], *elb = (const float*)d_in[23];

  // workspace carve
  char* w = (char*)d_ws;
  float* nf       = (float*)(w);                 // 8000*32 f32 = 1,024,000 B
  int*   rowcount = (int*)(w + 1024000);         // 8000 ints
  int*   rowstart = (int*)(w + 1056000);         // 8000 ints
  int*   nedge    = (int*)(w + 1088000);         // 1 int (padded)
  int*   eidx     = (int*)(w + 1088064);         // 400000*3 ints = 4.8 MB -> ends 5,888,064
  float* nW1p     = (float*)(w + 5888064);       // 32*128*4  = 16,384
  float* nW2p     = (float*)(w + 5904448);       // 64 KB
  float* nW3p     = (float*)(w + 5969984);       // 64 KB
  float* eW1p     = (float*)(w + 6035520);       // 4*128*4   = 2,048
  float* eW2p     = (float*)(w + 6037568);       // 64 KB
  float* eW3p     = (float*)(w + 6103104);       // 64 KB -> ends 6,168,640

  // output carve (flat f32, tuple order)
  float* out       = (float*)d_out;
  float* out_nodes = out;                        // 1,024,000
  float* out_mask  = out + 1024000;              //     8,000
  float* out_edges = out + 1032000;              // 51,200,000
  float* out_nidx  = out + 52232000;             //  1,200,000
  float* out_valid = out + 53432000;             //    400,000

  // one-shot weight packing (deterministic; cheap)
  pack_w_kernel<<<(16 * 128 + 255) / 256, 256, 0, stream>>>(nW1, 30, 32, nW1p);
  pack_w_kernel<<<(64 * 128 + 255) / 256, 256, 0, stream>>>(nW2, 128, 128, nW2p);
  pack_w_kernel<<<(64 * 128 + 255) / 256, 256, 0, stream>>>(nW3, 128, 128, nW3p);
  pack_w_kernel<<<(2 * 128 + 255) / 256, 256, 0, stream>>>(eW1, 3, 4, eW1p);
  pack_w_kernel<<<(64 * 128 + 255) / 256, 256, 0, stream>>>(eW2, 128, 128, eW2p);
  pack_w_kernel<<<(64 * 128 + 255) / 256, 256, 0, stream>>>(eW3, 128, 128, eW3p);

  node_feats_kernel<<<(NROWS + 255) / 256, 256, 0, stream>>>(pos, vel, mat, vmean, vstd,
                                                             matW, matb, nf);
  rowcount_kernel<<<NROWS / 4, 128, 0, stream>>>(pos, rowcount);
  scan_kernel<<<1, 1024, 0, stream>>>(rowcount, rowstart, nedge);
  emit_kernel<<<NROWS / 4, 128, 0, stream>>>(pos, rowstart, eidx);
  finalize_kernel<<<(MAXE + 255) / 256, 256, 0, stream>>>(eidx, nedge, out_nidx, out_valid);
  mask_kernel<<<(NROWS + 255) / 256, 256, 0, stream>>>(out_mask);
  node_mlp_kernel<<<(NROWS / 16) / WPB, WPB * 32, 0, stream>>>(
      nf, nW1p, nb1, nW2p, nb2, nW3p, nb3, nlg, nlb, out_nodes);
  edge_mlp_kernel<<<(MAXE / 16) / WPB, WPB * 32, 0, stream>>>(
      pos, eidx, nedge, eW1p, eb1, eW2p, eb2, eW3p, eb3, elg, elb, out_edges);
}